// DetLoss_87376814670265
// MI455X (gfx1250) — compile-verified
//
#include <hip/hip_runtime.h>
#include <math.h>

#define A_N 200000
#define B_N 8
#define G_N 16
#define NEG_N 8

typedef float v2f __attribute__((ext_vector_type(2)));
typedef float v8f __attribute__((ext_vector_type(8)));

// ws layout: ws[b*4 + q], q: 0=clf_sum, 1=sl1_pos_sum, 2=eiou_pos_sum, 3=num_pos
__global__ void det_loss_zero_ws(float* ws) {
    if (threadIdx.x < B_N * 4) ws[threadIdx.x] = 0.0f;
}

__global__ void __launch_bounds__(256) det_loss_main(
    const float*  __restrict__ clf,      // [B, A, 1]
    const float2* __restrict__ reg,      // [B, A, 2]
    const float2* __restrict__ anchors,  // [A, 2]
    const float2* __restrict__ gt,       // [B, G, 2]
    const float2* __restrict__ neg,      // [B, Nn, 2]
    float* __restrict__ ws)
{
    __shared__ float s_g0[G_N], s_g1[G_N], s_gw[G_N], s_gcx[G_N];
    __shared__ float s_n0[NEG_N], s_n1[NEG_N];
    __shared__ float s_red[8][4];

    const int b   = blockIdx.y;
    const int tid = threadIdx.x;

    if (tid < G_N) {
        float2 g = gt[b * G_N + tid];
        s_g0[tid]  = g.x;
        s_g1[tid]  = g.y;
        s_gw[tid]  = g.y - g.x;
        s_gcx[tid] = g.x + 0.5f * (g.y - g.x);
    }
    if (tid < NEG_N) {
        float2 nb = neg[b * NEG_N + tid];
        s_n0[tid] = nb.x;
        s_n1[tid] = nb.y;
    }
    __syncthreads();

    const float*  clf_b = clf + (size_t)b * A_N;
    const float2* reg_b = reg + (size_t)b * A_N;

    float clf_sum = 0.0f, sl1_sum = 0.0f, eiou_sum = 0.0f, npos = 0.0f;

    const int stride = gridDim.x * blockDim.x;
    const float BETA = 1.0f / 9.0f;

    for (int a = blockIdx.x * blockDim.x + tid; a < A_N; a += stride) {
        // prefetch next stride's stream data into cache (global_prefetch_b8)
        __builtin_prefetch(&clf_b[a + stride], 0, 1);
        __builtin_prefetch(&reg_b[a + stride], 0, 1);

        float2 an = anchors[a];
        float a0 = an.x, a1 = an.y;
        float aw  = a1 - a0;
        float acx = a0 + 0.5f * aw;

        // neg-anchor check
        bool neg_ind = false;
        #pragma unroll
        for (int j = 0; j < NEG_N; ++j) {
            float inter = fmaxf(fminf(a1, s_n1[j]) - fmaxf(a0, s_n0[j]), 0.0f);
            float uni   = aw + (s_n1[j] - s_n0[j]) - inter;
            neg_ind = neg_ind || (inter / uni > 0.75f);
        }

        // IoU vs GT, first-max argmax
        float best = -3.0e38f; int bi = 0;
        #pragma unroll
        for (int g = 0; g < G_N; ++g) {
            float inter = fmaxf(fminf(a1, s_g1[g]) - fmaxf(a0, s_g0[g]), 0.0f);
            float iou   = inter / (aw + s_gw[g] - inter);
            if (iou > best) { best = iou; bi = g; }
        }
        if (neg_ind) { best = -1.0f; bi = 0; }

        bool pos    = best >= 0.3f;
        bool ignore = (best >= 0.03f) && (best < 0.3f);
        float t = neg_ind ? 0.0f : (pos ? 1.0f : (ignore ? -1.0f : 0.0f));

        // focal classification loss
        float x   = clf_b[a];
        float bce = fmaxf(x, 0.0f) - x * t + log1pf(expf(-fabsf(x)));
        float p   = 1.0f / (1.0f + expf(-x));
        p = fminf(fmaxf(p, 1.0e-4f), 1.0f - 1.0e-4f);
        float alpha_f = pos ? 0.25f : 0.75f;
        float fw      = pos ? (1.0f - p) : p;
        float closs   = alpha_f * fw * fw * bce;
        if (ignore) closs = 0.0f;
        clf_sum += closs;

        float posf = pos ? 1.0f : 0.0f;
        npos += posf;

        // smooth-L1 on regression targets
        float2 r = reg_b[a];
        float gw  = s_gw[bi];
        float gcx = s_gcx[bi];
        float tr0 = ((gcx - acx) / aw) / 0.1f;
        float tr1 = logf(gw / aw) / 0.2f;
        float d0 = fabsf(tr0 - r.x);
        float d1 = fabsf(tr1 - r.y);
        float sl1 = ((d0 <= BETA) ? (0.5f * d0 * d0 / BETA) : (d0 - 0.5f * BETA))
                  + ((d1 <= BETA) ? (0.5f * d1 * d1 / BETA) : (d1 - 0.5f * BETA));
        sl1_sum += sl1 * posf;

        // EIoU
        float pred_ctr = acx + r.x * 0.1f * aw;
        float pred_w   = expf(r.y * 0.2f) * aw;
        float pb0 = fminf(fmaxf(pred_ctr - 0.5f * pred_w, 0.0f), 416.0f);
        float pb1 = fminf(fmaxf(pred_ctr + 0.5f * pred_w, 0.0f), 416.0f);
        float as0 = s_g0[bi], as1 = s_g1[bi];
        float inter = fmaxf(fminf(pb1, as1) - fmaxf(pb0, as0), 0.0f);
        float uni   = (pb1 - pb0) + (as1 - as0) - inter;
        float piou  = inter / uni;
        float dd    = fabsf(0.5f * (pb0 + pb1) - 0.5f * (as0 + as1));
        float c     = fmaxf(pb1, as1) - fminf(pb0, as0);
        float c2    = fmaxf(c * c, 1.0e-6f);
        float wd    = fabsf((pb1 - pb0) - (as1 - as0));
        float el    = 1.0f - (piou - dd * dd / c2 - wd * wd / c2);
        eiou_sum += el * posf;
    }

    // wave32 reduction via shuffles
    #pragma unroll
    for (int off = 16; off >= 1; off >>= 1) {
        clf_sum  += __shfl_down(clf_sum,  off, 32);
        sl1_sum  += __shfl_down(sl1_sum,  off, 32);
        eiou_sum += __shfl_down(eiou_sum, off, 32);
        npos     += __shfl_down(npos,     off, 32);
    }
    const int wave = tid >> 5;
    const int lane = tid & 31;
    if (lane == 0) {
        s_red[wave][0] = clf_sum;
        s_red[wave][1] = sl1_sum;
        s_red[wave][2] = eiou_sum;
        s_red[wave][3] = npos;
    }
    __syncthreads();

    // Block-level reduction of 8 waves x 4 quantities via one f32 WMMA:
    // D = ones(16x4) x B(4x16), i.e. column sums of B. Column n (n<8):
    //   q = n&3 (quantity), g = n>>2 (wave half); colsum_n = sum_k V[g*4+k][q].
    // S_q = colsum_q + colsum_{q+4}. Exact f32 arithmetic.
    if (tid < 32) {  // whole wave 0 -> EXEC all ones inside (WMMA requirement)
        int n = lane & 15;
        int q = n & 3;
        int g = (n >> 2) & 3;
        int k = lane >> 4;            // B row half: vgpr0 -> k or vgpr1 -> k+2
        float b0v = 0.0f, b1v = 0.0f;
        if (n < 8) {
            b0v = s_red[g * 4 + k][q];
            b1v = s_red[g * 4 + k + 2][q];
        }
        v2f av; av[0] = 1.0f; av[1] = 1.0f;   // A = ones(16x4)
        v2f bv; bv[0] = b0v;  bv[1] = b1v;
        v8f cv = {};
        v8f dv = __builtin_amdgcn_wmma_f32_16x16x4_f32(
            /*neg_a=*/false, av, /*neg_b=*/false, bv,
            /*c_mod=*/(short)0, cv, /*reuse_a=*/false, /*reuse_b=*/false);
        float d0 = dv[0];                               // D[0, n] for lanes 0..15
        float other = __shfl(d0, (lane & 3) + 4, 32);   // colsum_{q+4}
        if (lane < 4) {
            atomicAdd(&ws[b * 4 + lane], d0 + other);
        }
    }
}

__global__ void det_loss_finalize(const float* __restrict__ ws,
                                  float* __restrict__ out)
{
    __shared__ float s_clf[B_N], s_reg[B_N];
    const int tid = threadIdx.x;
    if (tid < B_N) {
        float csum = ws[tid * 4 + 0];
        float ssum = ws[tid * 4 + 1];
        float esum = ws[tid * 4 + 2];
        float np   = ws[tid * 4 + 3];
        float denom     = fmaxf(np, 1.0f);
        float clf_total = csum / denom;
        float reg_mean  = ssum / (denom * 2.0f);
        float eiou_mean = esum / denom;
        float reg_total = (np > 0.0f) ? (1.0f * reg_mean + 1.5f * eiou_mean) : 0.0f;
        s_clf[tid] = clf_total;
        s_reg[tid] = reg_total;
    }
    __syncthreads();
    if (tid == 0) {
        float c = 0.0f, r = 0.0f;
        #pragma unroll
        for (int i = 0; i < B_N; ++i) { c += s_clf[i]; r += s_reg[i]; }
        out[0] = c * (1.0f / B_N);
        out[1] = r * (1.0f / B_N);
    }
}

extern "C" void kernel_launch(void* const* d_in, const int* in_sizes, int n_in,
                              void* d_out, int out_size, void* d_ws, size_t ws_size,
                              hipStream_t stream) {
    const float*  clf     = (const float*) d_in[0];
    const float2* reg     = (const float2*)d_in[1];
    const float2* anchors = (const float2*)d_in[2];
    const float2* gt      = (const float2*)d_in[3];
    const float2* neg     = (const float2*)d_in[4];
    float* ws  = (float*)d_ws;
    float* out = (float*)d_out;

    det_loss_zero_ws<<<1, 64, 0, stream>>>(ws);

    // ~2 anchors per thread per batch-item; plenty of blocks to fill the chip.
    const int blocks_x = (A_N + 512 - 1) / 512;   // 391
    dim3 grid(blocks_x, B_N, 1);
    det_loss_main<<<grid, 256, 0, stream>>>(clf, reg, anchors, gt, neg, ws);

    det_loss_finalize<<<1, 32, 0, stream>>>(ws, out);
}